// HeatMap_2061584302326
// MI455X (gfx1250) — compile-verified
//
#include <hip/hip_runtime.h>

// MI455X / gfx1250 fused kernel:
//   scores[b,q,t,hw,g] = sum_c dec[b,q,t,g,c] * enc[b,t,hw,g,c]   (WMMA f32 16x16x4, K=64)
//   out[b,q,t,hw,f]    = relu(sum_g scores * W[g,f] + bias[f])    (WMMA f32 16x16x4, K=8)
// Store-bound: 256 MB fp32 output @ 23.3 TB/s ~ 11 us floor; WMMA keeps compute far under that.
// enc chunk is staged global->LDS via the CDNA5 async-copy path (ASYNCcnt, no VGPR staging).

typedef __attribute__((ext_vector_type(2))) float v2f;
typedef __attribute__((ext_vector_type(4))) float v4f;
typedef __attribute__((ext_vector_type(8))) float v8f;

#define B_      4
#define QP_     20   // dec has q+4 = 20, we use first 16
#define Q_      16
#define T_      8
#define HW_     256
#define D_      512
#define G_      8
#define C_      64
#define F_      512

#define HWCHUNK 32          // hw positions per block -> 4*8*8 = 256 blocks
// ---- LDS layout (floats), dynamic shared ----
#define ENC_STRIDE 516      // 512 + pad(4): keeps b128 16B-aligned, spreads banks (4*lane)
#define ENC_FLOATS (HWCHUNK * ENC_STRIDE)          // 16512
#define WT_STRIDE  10       // Wt[f][k], pad to 10: conflict-free b64 reads (10*lane % 64 distinct)
#define WT_OFF     ENC_FLOATS
#define WT_FLOATS  (F_ * WT_STRIDE)                // 5120
#define SC_STRIDE  10       // scores[m][g], same padding rationale
#define SC_OFF     (WT_OFF + WT_FLOATS)
#define SC_FLOATS  (Q_ * HWCHUNK * SC_STRIDE)      // 5120
#define LDS_FLOATS (SC_OFF + SC_FLOATS)            // 26752 floats = 107008 B

__global__ __launch_bounds__(256, 1)
void heatmap_fused(const float* __restrict__ dec,   // (4,20,8,512)
                   const float* __restrict__ enc,   // (4,8,16,16,512) -> (4,8,256,512)
                   const float* __restrict__ Wm,    // (8,512)
                   const float* __restrict__ bias,  // (512)
                   float* __restrict__ out)         // (4,16,8,256,512)
{
    extern __shared__ float smem[];
    float* encS = smem;            // [HWCHUNK][ENC_STRIDE]
    float* WtS  = smem + WT_OFF;   // [F][WT_STRIDE]  Wt[f][k] = W[k][f]
    float* scS  = smem + SC_OFF;   // [Q*HWCHUNK][SC_STRIDE]

    const int tid  = threadIdx.x;
    const int wave = tid >> 5;     // 0..7  (wave32)
    const int lane = tid & 31;
    const int ln   = lane & 15;    // lane % 16
    const int hi   = lane >> 4;    // half-wave select

    const int blk   = blockIdx.x;          // 0..255
    const int b     = blk >> 6;            // 0..3
    const int t     = (blk >> 3) & 7;      // 0..7
    const int chunk = blk & 7;             // 0..7
    const int hw0   = chunk * HWCHUNK;

    // ---------- staging ----------
    // enc chunk -> encS via async global->LDS copies (ASYNCcnt path, bypasses VGPRs).
    // Each thread moves 64 floats = 16 x b128. LDS dest keeps the padded ENC_STRIDE layout.
    {
        const float* encBase = enc + (((size_t)(b * T_ + t)) * HW_ + hw0) * D_;
        const int r     = tid >> 3;         // row 0..31 (8 threads per 512-float row)
        const int cbase = (tid & 7) * 64;   // 64 floats per thread
        #pragma unroll
        for (int i = 0; i < 16; ++i) {
            // Generic->LDS: flat aperture truncates to addr[31:0] = LDS byte offset.
            unsigned int ldsAddr =
                (unsigned int)(uintptr_t)(encS + r * ENC_STRIDE + cbase + i * 4);
            unsigned long long gAddr =
                (unsigned long long)(uintptr_t)(encBase + (size_t)r * D_ + cbase + i * 4);
            asm volatile("global_load_async_to_lds_b128 %0, %1, off"
                         :: "v"(ldsAddr), "v"(gAddr) : "memory");
        }
        // W -> Wt[f][k] transpose (needs VGPR pass; 16KB, coalesced reads).
        #pragma unroll
        for (int i = 0; i < 16; ++i) {      // 4096 W elements
            const int idx = tid + i * 256;
            const int k   = idx >> 9;       // 0..7
            const int f   = idx & 511;
            WtS[f * WT_STRIDE + k] = Wm[idx];
        }
        asm volatile("s_wait_asynccnt 0x0" ::: "memory");  // async copies done before barrier
    }
    __syncthreads();

    // ---------- stage 1: wave w computes head g = w ----------
    // scores(q,hw) = dec(q, c) x enc(hw, c)^T  : 16x32x64 GEMM = 2 hw-tiles x 16 k-steps
    {
        const int g = wave;
        // A operand preload: A[M=ln][K = ks*4 + 2*hi + {0,1}] from global (L2-hot, 1.3MB array)
        const float* decRow = dec + (((size_t)(b * QP_ + ln)) * T_ + t) * D_ + g * C_ + 2 * hi;
        v2f a[16];
        #pragma unroll
        for (int ks = 0; ks < 16; ++ks)
            a[ks] = *(const v2f*)(decRow + ks * 4);

        #pragma unroll
        for (int hwt = 0; hwt < 2; ++hwt) {
            const int hwn = hwt * 16 + ln;                       // B: N = hwn
            const float* bRow = encS + hwn * ENC_STRIDE + g * C_ + 2 * hi;
            v8f c = {};
            #pragma unroll
            for (int ks = 0; ks < 16; ++ks) {
                v2f bv = *(const v2f*)(bRow + ks * 4);           // B[K=c][N=hw] = enc[hw][c]
                c = __builtin_amdgcn_wmma_f32_16x16x4_f32(
                        false, a[ks], false, bv, (short)0, c, false, false);
            }
            // D layout: VGPR v -> row q = v + 8*hi, col hw = hwn. Scatter into scores[m][g].
            #pragma unroll
            for (int v = 0; v < 8; ++v) {
                const int m = (v + 8 * hi) * HWCHUNK + hwn;      // m = q*32 + hw_local
                scS[m * SC_STRIDE + g] = c[v];
            }
        }
    }
    __syncthreads();

    // ---------- stage 2: out = relu(scores(m,g) x W(g,f) + bias) ----------
    // M = 512 (32 m-tiles), N = 512 (32 n-tiles), K = 8 (2 wmma k-steps). Wave owns 4 m-tiles.
    {
        v2f a[4][2];
        #pragma unroll
        for (int mi = 0; mi < 4; ++mi) {
            const int m = (wave * 4 + mi) * 16 + ln;
            #pragma unroll
            for (int s = 0; s < 2; ++s)
                a[mi][s] = *(const v2f*)(scS + m * SC_STRIDE + s * 4 + 2 * hi);
        }
        for (int nt = 0; nt < 32; ++nt) {
            const int f = nt * 16 + ln;
            const v2f w0 = *(const v2f*)(WtS + f * WT_STRIDE + 0 + 2 * hi);  // K = 0..3 slice
            const v2f w1 = *(const v2f*)(WtS + f * WT_STRIDE + 4 + 2 * hi);  // K = 4..7 slice
            const float bv = bias[f];
            #pragma unroll
            for (int mi = 0; mi < 4; ++mi) {
                const int mt = wave * 4 + mi;
                v8f c = {};
                c = __builtin_amdgcn_wmma_f32_16x16x4_f32(false, a[mi][0], false, w0, (short)0, c, false, false);
                c = __builtin_amdgcn_wmma_f32_16x16x4_f32(false, a[mi][1], false, w1, (short)0, c, false, false);
                // D element v -> m = mt*16 + 8*hi + v -> q = mt>>1, hwl = (mt&1)*16 + 8*hi + v
                const int q = mt >> 1;
                const int hwlBase = (mt & 1) * 16 + 8 * hi;
                float* outRow = out + ((((size_t)(b * Q_ + q)) * T_ + t) * HW_ + hw0 + hwlBase) * F_ + f;
                #pragma unroll
                for (int v = 0; v < 8; ++v) {
                    float o = c[v] + bv;
                    o = o > 0.f ? o : 0.f;                       // ReLU
                    outRow[(size_t)v * F_] = o;                  // 64B-contiguous per half-wave
                }
            }
        }
    }
}

extern "C" void kernel_launch(void* const* d_in, const int* in_sizes, int n_in,
                              void* d_out, int out_size, void* d_ws, size_t ws_size,
                              hipStream_t stream) {
    (void)in_sizes; (void)n_in; (void)out_size; (void)d_ws; (void)ws_size;
    const float* dec  = (const float*)d_in[0];
    const float* enc  = (const float*)d_in[1];
    const float* Wm   = (const float*)d_in[2];
    const float* bias = (const float*)d_in[3];
    float* out = (float*)d_out;

    const size_t shmem = (size_t)LDS_FLOATS * sizeof(float);   // ~104.5 KB dynamic LDS
    heatmap_fused<<<dim3(256), dim3(256), shmem, stream>>>(dec, enc, Wm, bias, out);
}